// HS_MSA_29523605192892
// MI455X (gfx1250) — compile-verified
//
#include <hip/hip_runtime.h>
#include <math.h>

// ---------------- constants ----------------
#define HH 256
#define WW 320
#define CC 28
#define NHEADS 8
#define DH 28
#define IDIM 224         // NHEADS*DH
#define HW 81920         // HH*WW
#define PI_D 3.14159265358979323846

typedef __attribute__((ext_vector_type(2))) float v2f;
typedef __attribute__((ext_vector_type(8))) float v8f;

__device__ __forceinline__ float gelu_exact(float x) {
    return 0.5f * x * (1.0f + erff(x * 0.70710678118654752440f));
}

// ---------------- DCT matrix generation ----------------
// D[k,n] = 2*cos(pi*(2n+1)k/(2N));  DI = D^T * diag(1/(4N), 1/(2N), ...)
__global__ void gen_dct_kernel(float* __restrict__ D, float* __restrict__ DI, int N) {
    int i = blockIdx.x * blockDim.x + threadIdx.x;
    if (i >= N * N) return;
    int k = i / N, n = i % N;
    double v = 2.0 * cos(PI_D * (2.0 * (double)n + 1.0) * (double)k / (2.0 * (double)N));
    D[(size_t)k * N + n] = (float)v;
    double s = (k == 0) ? 1.0 / (4.0 * (double)N) : 1.0 / (2.0 * (double)N);
    DI[(size_t)n * N + k] = (float)(v * s);
}

// ---------------- generic per-image 256x320 WMMA f32 GEMM ----------------
// C[b,c][m,n] = sum_k A[b,c][m,k] * B[b,c][k,n]   (A row-contiguous in k: aCol==1)
// One wave computes a 16x64 strip: 4 independent v8f accumulators sharing the
// A fragment (2.5 loads/lane per WMMA instead of 4; WMMAs per K-step independent).
// tilesM=16 (M=256), tilesN=5 x 64 (N=320), 56 images -> 4480 waves.
// Layouts per CDNA5 ISA 7.12.2:
//   A 16x4 : lane&15 = M row, K = k0 + 2*(lane>>4) + {vgpr0,1}
//   B 4x16 : lane&15 = N col, K = k0 + 2*(lane>>4) + {vgpr0,1}
//   C/D    : lane&15 = N col, M row = m0 + vgpr + 8*(lane>>4)
template <bool BK_CONTIG>
__global__ void __launch_bounds__(32)
wmma_gemm_img(const float* __restrict__ A, const float* __restrict__ B,
              float* __restrict__ Cm, int KK,
              int aOffB, int aOffC, int aRow,
              int bOffB, int bOffC, int bRow, int bCol,
              int cOffB, int cOffC) {
    int blk = blockIdx.x;
    int nt = blk % 5;  blk /= 5;
    int mt = blk % 16; blk /= 16;
    int c = blk % CC;
    int b = blk / CC;
    int lane = threadIdx.x;
    int r  = lane & 15;
    int hf = lane >> 4;
    int m0 = mt * 16, n0 = nt * 64;
    const float* Ab = A + (size_t)b * aOffB + (size_t)c * aOffC;
    const float* Bb = B + (size_t)b * bOffB + (size_t)c * bOffC;
    v8f z = {0.f, 0.f, 0.f, 0.f, 0.f, 0.f, 0.f, 0.f};
    v8f a0 = z, a1 = z, a2 = z, a3 = z;
    for (int k0 = 0; k0 < KK; k0 += 4) {
        int ka = k0 + 2 * hf;                       // even -> 8B-aligned v2f loads
        v2f av = *(const v2f*)(Ab + (size_t)(m0 + r) * aRow + ka);
        v2f bv0, bv1, bv2, bv3;
        if (BK_CONTIG) {                            // B[k][n] = Bb[k + n*bCol]
            bv0 = *(const v2f*)(Bb + ka + (size_t)(n0 + r)      * bCol);
            bv1 = *(const v2f*)(Bb + ka + (size_t)(n0 + 16 + r) * bCol);
            bv2 = *(const v2f*)(Bb + ka + (size_t)(n0 + 32 + r) * bCol);
            bv3 = *(const v2f*)(Bb + ka + (size_t)(n0 + 48 + r) * bCol);
        } else {                                    // B[k][n] = Bb[k*bRow + n*bCol]
            const float* b0p = Bb + (size_t)ka * bRow;
            const float* b1p = b0p + bRow;
            bv0.x = b0p[(size_t)(n0 + r)      * bCol]; bv0.y = b1p[(size_t)(n0 + r)      * bCol];
            bv1.x = b0p[(size_t)(n0 + 16 + r) * bCol]; bv1.y = b1p[(size_t)(n0 + 16 + r) * bCol];
            bv2.x = b0p[(size_t)(n0 + 32 + r) * bCol]; bv2.y = b1p[(size_t)(n0 + 32 + r) * bCol];
            bv3.x = b0p[(size_t)(n0 + 48 + r) * bCol]; bv3.y = b1p[(size_t)(n0 + 48 + r) * bCol];
        }
        a0 = __builtin_amdgcn_wmma_f32_16x16x4_f32(false, av, false, bv0, (short)0, a0, false, false);
        a1 = __builtin_amdgcn_wmma_f32_16x16x4_f32(false, av, false, bv1, (short)0, a1, false, false);
        a2 = __builtin_amdgcn_wmma_f32_16x16x4_f32(false, av, false, bv2, (short)0, a2, false, false);
        a3 = __builtin_amdgcn_wmma_f32_16x16x4_f32(false, av, false, bv3, (short)0, a3, false, false);
    }
    float* Cb = Cm + (size_t)b * cOffB + (size_t)c * cOffC;
#pragma unroll
    for (int v = 0; v < 8; ++v) {
        size_t rowb = (size_t)(m0 + v + 8 * hf) * WW + n0 + r;
        Cb[rowb]      = a0[v];
        Cb[rowb + 16] = a1[v];
        Cb[rowb + 32] = a2[v];
        Cb[rowb + 48] = a3[v];
    }
}

// ---------------- spectral 8x8-patch channel attention ----------------
// One block per (b, patch). XD [b,c,H,W] -> XLOW [b,c,H,W]
#define SPT 256
__global__ void __launch_bounds__(SPT)
spec_attn_kernel(const float* __restrict__ XD, float* __restrict__ XLOW,
                 const float* __restrict__ wq, const float* __restrict__ wk,
                 const float* __restrict__ wv, const float* __restrict__ rescale,
                 const float* __restrict__ projw, const float* __restrict__ projb) {
    __shared__ float xt[64][28];
    __shared__ float qs[28][64];
    __shared__ float ks[28][64];
    __shared__ float vs[28][64];
    __shared__ float at[28][28];
    __shared__ float accm[64][28];
    int wid = blockIdx.x;           // 0..2559
    int b = wid / 1280;
    int t = wid % 1280;
    int ph = t / 40, pw = t % 40;   // hh=32, ww=40
    int tid = threadIdx.x;
    for (int idx = tid; idx < 64 * 28; idx += SPT) {
        int m = idx / 28, ci = idx % 28;
        int i = m >> 3, j = m & 7;
        xt[m][ci] = XD[(((size_t)b * CC + ci) * HH + ph * 8 + i) * WW + pw * 8 + j];
        accm[m][ci] = 0.f;
    }
    __syncthreads();
    for (int h = 0; h < NHEADS; ++h) {
        // q/k/v projections -> [d][m]
        for (int idx = tid; idx < 28 * 64; idx += SPT) {
            int d = idx / 64, m = idx % 64;
            int col = h * DH + d;
            float aq = 0.f, ak = 0.f, av = 0.f;
            for (int ci = 0; ci < 28; ++ci) {
                float xv = xt[m][ci];
                aq += xv * wq[ci * IDIM + col];
                ak += xv * wk[ci * IDIM + col];
                av += xv * wv[ci * IDIM + col];
            }
            qs[d][m] = aq; ks[d][m] = ak; vs[d][m] = av;
        }
        __syncthreads();
        // L2-normalize q and k rows over m=64
        for (int rowi = tid; rowi < 56; rowi += SPT) {
            float* row = (rowi < 28) ? qs[rowi] : ks[rowi - 28];
            float s = 0.f;
            for (int m = 0; m < 64; ++m) s += row[m] * row[m];
            float inv = 1.0f / fmaxf(sqrtf(s), 1e-12f);
            for (int m = 0; m < 64; ++m) row[m] *= inv;
        }
        __syncthreads();
        float rs = rescale[h];
        for (int idx = tid; idx < 28 * 28; idx += SPT) {
            int d = idx / 28, e = idx % 28;
            float s = 0.f;
            for (int m = 0; m < 64; ++m) s += qs[d][m] * ks[e][m];
            at[d][e] = s * rs;
        }
        __syncthreads();
        // softmax rows over e
        for (int d = tid; d < 28; d += SPT) {
            float mx = -1e30f;
            for (int e = 0; e < 28; ++e) mx = fmaxf(mx, at[d][e]);
            float sm = 0.f;
            for (int e = 0; e < 28; ++e) { float ev = expf(at[d][e] - mx); at[d][e] = ev; sm += ev; }
            float inv = 1.0f / sm;
            for (int e = 0; e < 28; ++e) at[d][e] *= inv;
        }
        __syncthreads();
        // x0 = attn @ v  (overwrite qs)
        for (int idx = tid; idx < 28 * 64; idx += SPT) {
            int d = idx / 64, m = idx % 64;
            float s = 0.f;
            for (int e = 0; e < 28; ++e) s += at[d][e] * vs[e][m];
            qs[d][m] = s;
        }
        __syncthreads();
        // out-projection partial over this head's id range
        for (int idx = tid; idx < 64 * 28; idx += SPT) {
            int m = idx / 28, ci = idx % 28;
            float s = 0.f;
            for (int d = 0; d < 28; ++d) s += qs[d][m] * projw[(size_t)(h * DH + d) * CC + ci];
            accm[m][ci] += s;
        }
        __syncthreads();
    }
    for (int idx = tid; idx < 64 * 28; idx += SPT) {
        int m = idx / 28, ci = idx % 28;
        int i = m >> 3, j = m & 7;
        XLOW[(((size_t)b * CC + ci) * HH + ph * 8 + i) * WW + pw * 8 + j] = accm[m][ci] + projb[ci];
    }
}

// ---------------- HF path: pointwise conv 1 (gelu) ----------------
__global__ void hf_pw1_kernel(const float* __restrict__ XD, const float* __restrict__ pw1,
                              float* __restrict__ Y) {
    int idx = blockIdx.x * blockDim.x + threadIdx.x;   // (b, h, w)
    if (idx >= 2 * HW) return;
    int b = idx / HW, hw = idx % HW;
    float xin[28];
    for (int ci = 0; ci < 28; ++ci) xin[ci] = XD[((size_t)b * CC + ci) * HW + hw];
    for (int co = 0; co < 28; ++co) {
        float s = 0.f;
        for (int ci = 0; ci < 28; ++ci) s += pw1[co * 28 + ci] * xin[ci];
        Y[((size_t)b * CC + co) * HW + hw] = gelu_exact(s);
    }
}

// ---------------- HF path: dw3x3 + pw2 + final spectral combine (in-place) ----------------
__global__ void hf_combine_kernel(const float* __restrict__ XD, const float* __restrict__ Y,
                                  float* __restrict__ XLO /* in: XLOW, out: XOUT */,
                                  const float* __restrict__ dw, const float* __restrict__ pw2,
                                  const float* __restrict__ coef) {
    int idx = blockIdx.x * blockDim.x + threadIdx.x;
    if (idx >= 2 * HW) return;
    int b = idx / HW, hw = idx % HW;
    int hh = hw / WW, wp = hw % WW;
    float xc[28];
    for (int ci = 0; ci < 28; ++ci) {
        float s = 0.f;
        const float* Yc = Y + ((size_t)b * CC + ci) * HW;
        for (int di = -1; di <= 1; ++di) {
            int hy = hh + di;
            if (hy < 0 || hy >= HH) continue;
            for (int dj = -1; dj <= 1; ++dj) {
                int wx = wp + dj;
                if (wx < 0 || wx >= WW) continue;
                s += dw[ci * 9 + (di + 1) * 3 + (dj + 1)] * Yc[hy * WW + wx];
            }
        }
        xc[ci] = gelu_exact(s) + XD[((size_t)b * CC + ci) * HW + hw];
    }
    float cf = coef[hw];
    for (int co = 0; co < 28; ++co) {
        float s = 0.f;
        for (int ci = 0; ci < 28; ++ci) s += pw2[co * 28 + ci] * xc[ci];
        float xhigh = gelu_exact(s) + xc[co];
        size_t o = ((size_t)b * CC + co) * HW + hw;
        float xlow = XLO[o];
        XLO[o] = cf * xlow + (1.0f - cf) * xhigh + XD[o];
    }
}

// ---------------- local 8x8 window attention ----------------
#define LPT 256
__global__ void __launch_bounds__(LPT)
local_attn_kernel(const float* __restrict__ X, float* __restrict__ OUT,
                  const float* __restrict__ wq, const float* __restrict__ wkv,
                  const float* __restrict__ pos, const float* __restrict__ projw,
                  const float* __restrict__ projb) {
    __shared__ float xw[64][28];
    __shared__ float qb[64][28];
    __shared__ float kb[64][28];
    __shared__ float vb[64][28];
    __shared__ float sim[64][64];
    __shared__ float accm[64][28];
    int wid = blockIdx.x;
    int b = wid / 1280;
    int t = wid % 1280;
    int wh = t / 40, wn = t % 40;   // nh=32, nw=40
    int tid = threadIdx.x;
    for (int idx = tid; idx < 64 * 28; idx += LPT) {
        int m = idx / 28, ci = idx % 28;
        int i = m >> 3, j = m & 7;
        xw[m][ci] = X[(((size_t)b * HH + wh * 8 + i) * WW + wn * 8 + j) * CC + ci];
        accm[m][ci] = 0.f;
    }
    __syncthreads();
    const float qscale = 0.18898223650461363f;  // 28^-0.5
    for (int h = 0; h < NHEADS; ++h) {
        for (int idx = tid; idx < 64 * 28; idx += LPT) {
            int i = idx / 28, d = idx % 28;
            float aq = 0.f, ak = 0.f, av = 0.f;
            int col = h * DH + d;
            for (int ci = 0; ci < 28; ++ci) {
                float xv = xw[i][ci];
                aq += xv * wq[ci * IDIM + col];
                ak += xv * wkv[ci * 2 * IDIM + col];
                av += xv * wkv[ci * 2 * IDIM + IDIM + col];
            }
            qb[i][d] = aq * qscale; kb[i][d] = ak; vb[i][d] = av;
        }
        __syncthreads();
        for (int idx = tid; idx < 64 * 64; idx += LPT) {
            int i = idx / 64, j = idx % 64;
            float s = 0.f;
            for (int d = 0; d < 28; ++d) s += qb[i][d] * kb[j][d];
            sim[i][j] = s + pos[((size_t)h * 64 + i) * 64 + j];
        }
        __syncthreads();
        for (int i = tid; i < 64; i += LPT) {
            float mx = -1e30f;
            for (int j = 0; j < 64; ++j) mx = fmaxf(mx, sim[i][j]);
            float sm = 0.f;
            for (int j = 0; j < 64; ++j) { float e = expf(sim[i][j] - mx); sim[i][j] = e; sm += e; }
            float inv = 1.0f / sm;
            for (int j = 0; j < 64; ++j) sim[i][j] *= inv;
        }
        __syncthreads();
        // o = a @ v  (overwrite qb)
        for (int idx = tid; idx < 64 * 28; idx += LPT) {
            int i = idx / 28, d = idx % 28;
            float s = 0.f;
            for (int j = 0; j < 64; ++j) s += sim[i][j] * vb[j][d];
            qb[i][d] = s;
        }
        __syncthreads();
        for (int idx = tid; idx < 64 * 28; idx += LPT) {
            int i = idx / 28, ci = idx % 28;
            float s = 0.f;
            for (int d = 0; d < 28; ++d) s += qb[i][d] * projw[(size_t)(h * DH + d) * CC + ci];
            accm[i][ci] += s;
        }
        __syncthreads();
    }
    for (int idx = tid; idx < 64 * 28; idx += LPT) {
        int m = idx / 28, ci = idx % 28;
        int i = m >> 3, j = m & 7;
        OUT[(((size_t)b * HH + wh * 8 + i) * WW + wn * 8 + j) * CC + ci] = accm[m][ci] + projb[ci];
    }
}

// ---------------- fusion: out = fus_w * [out_fd ; out_local] + fus_b ----------------
// out_local already lives in OUT; per-pixel read-then-write within one thread.
__global__ void fusion_kernel(const float* __restrict__ FD, float* __restrict__ OUT,
                              const float* __restrict__ fw, const float* __restrict__ fb) {
    int idx = blockIdx.x * blockDim.x + threadIdx.x;
    if (idx >= 2 * HW) return;
    int b = idx / HW, hw = idx % HW;
    float lf[28], ll[28];
    for (int ci = 0; ci < 28; ++ci) {
        lf[ci] = FD[((size_t)b * CC + ci) * HW + hw];   // out_fd [b,c,h,w]
        ll[ci] = OUT[(size_t)idx * CC + ci];            // out_local [b,h,w,c]
    }
    for (int o = 0; o < 28; ++o) {
        float s = fb[o];
        for (int ci = 0; ci < 28; ++ci)
            s += fw[o * 56 + ci] * lf[ci] + fw[o * 56 + 28 + ci] * ll[ci];
        OUT[(size_t)idx * CC + o] = s;
    }
}

// ---------------- host launch ----------------
extern "C" void kernel_launch(void* const* d_in, const int* in_sizes, int n_in,
                              void* d_out, int out_size, void* d_ws, size_t ws_size,
                              hipStream_t stream) {
    const float* x            = (const float*)d_in[0];
    const float* spec_wq      = (const float*)d_in[1];
    const float* spec_wk      = (const float*)d_in[2];
    const float* spec_wv      = (const float*)d_in[3];
    const float* spec_rescale = (const float*)d_in[4];
    const float* spec_proj_w  = (const float*)d_in[5];
    const float* spec_proj_b  = (const float*)d_in[6];
    const float* hf1_pw_w     = (const float*)d_in[7];
    const float* hf1_dw_w     = (const float*)d_in[8];
    const float* hf2_pw_w     = (const float*)d_in[9];
    const float* coef_emb     = (const float*)d_in[10];
    const float* loc_wq       = (const float*)d_in[11];
    const float* loc_wkv      = (const float*)d_in[12];
    const float* loc_pos      = (const float*)d_in[13];
    const float* loc_proj_w   = (const float*)d_in[14];
    const float* loc_proj_b   = (const float*)d_in[15];
    const float* fus_w        = (const float*)d_in[16];
    const float* fus_b        = (const float*)d_in[17];
    float* out = (float*)d_out;

    const size_t IMG = (size_t)2 * CC * HW;             // 4,587,520 floats per staging buffer
    const size_t WS_FLOATS = 3 * IMG + 2 * 256 * 256 + 2 * 320 * 320;
    if (ws_size < WS_FLOATS * sizeof(float)) return;    // deterministic guard

    float* B0  = (float*)d_ws;          // T1 -> Y -> T2
    float* B1  = B0 + IMG;              // XD -> OUTFD
    float* B2  = B1 + IMG;              // XLOW -> XOUT (in place)
    float* DH_ = B2 + IMG;
    float* DIH = DH_ + 256 * 256;
    float* DW_ = DIH + 256 * 256;
    float* DIW = DW_ + 320 * 320;

    // 0) generate DCT / inverse-DCT matrices (module constants in the reference)
    gen_dct_kernel<<<(256 * 256 + 255) / 256, 256, 0, stream>>>(DH_, DIH, 256);
    gen_dct_kernel<<<(320 * 320 + 255) / 256, 256, 0, stream>>>(DW_, DIW, 320);

    const int WC = WW * CC;             // 8960
    const int nGemmBlk = 56 * 16 * 5;   // 4480 waves, one 16x64 strip each

    // 1) DCT rows: T1[b,c,k,w] = sum_h DH[k,h] * x[b,h,w,c]       (B strided in k)
    wmma_gemm_img<false><<<nGemmBlk, 32, 0, stream>>>(DH_, x, B0, 256,
        /*A*/ 0, 0, 256,
        /*B*/ HW * CC, 1, WC, CC,
        /*C*/ CC * HW, HW);
    // 2) DCT cols: XD[b,c,k,l] = sum_w T1[k,w] * DW[l,w]          (B contiguous in k)
    wmma_gemm_img<true><<<nGemmBlk, 32, 0, stream>>>(B0, DW_, B1, 320,
        /*A*/ CC * HW, HW, WW,
        /*B*/ 0, 0, 1, 320,
        /*C*/ CC * HW, HW);
    // 3) spectral patch attention: XD -> XLOW
    spec_attn_kernel<<<2560, SPT, 0, stream>>>(B1, B2, spec_wq, spec_wk, spec_wv,
                                               spec_rescale, spec_proj_w, spec_proj_b);
    // 4) HF pointwise-1: XD -> Y (B0, T1 dead)
    hf_pw1_kernel<<<(2 * HW + 255) / 256, 256, 0, stream>>>(B1, hf1_pw_w, B0);
    // 5) dw3x3 + pw2 + combine -> XOUT in place in B2
    hf_combine_kernel<<<(2 * HW + 255) / 256, 256, 0, stream>>>(B1, B0, B2,
                                                                hf1_dw_w, hf2_pw_w, coef_emb);
    // 6) IDCT rows: T2[b,c,k,w] = sum_h DIH[k,h] * XOUT[b,c,h,w]  (B2 -> B0, B strided in k)
    wmma_gemm_img<false><<<nGemmBlk, 32, 0, stream>>>(DIH, B2, B0, 256,
        /*A*/ 0, 0, 256,
        /*B*/ CC * HW, HW, WW, 1,
        /*C*/ CC * HW, HW);
    // 7) IDCT cols: OUTFD[b,c,k,l] = sum_w T2[k,w] * DIW[l,w]     (B0 -> B1, B contiguous in k)
    wmma_gemm_img<true><<<nGemmBlk, 32, 0, stream>>>(B0, DIW, B1, 320,
        /*A*/ CC * HW, HW, WW,
        /*B*/ 0, 0, 1, 320,
        /*C*/ CC * HW, HW);
    // 8) local window attention -> d_out (out_local, [b,h,w,c])
    local_attn_kernel<<<2560, LPT, 0, stream>>>(x, out, loc_wq, loc_wkv, loc_pos,
                                                loc_proj_w, loc_proj_b);
    // 9) fusion: d_out = fus_w * [OUTFD ; d_out] + fus_b
    fusion_kernel<<<(2 * HW + 255) / 256, 256, 0, stream>>>(B1, out, fus_w, fus_b);
}